// cross_entropy_margin_83451214561473
// MI455X (gfx1250) — compile-verified
//
#include <hip/hip_runtime.h>
#include <hip/hip_bf16.h>

// Margin cross-entropy: loss = -mean_i [ 4*l_y - log( sum_j exp(l_j/4) - exp(l_y/4) + exp(4*l_y) ) ]
// Bandwidth-bound: 1.05 GB streamed once; floor ~45us @ 23.3 TB/s.

#define CEM_BATCH 8192
#define CEM_CLASSES 32000
#define CEM_CLASSES4 (CEM_CLASSES / 4)
#define CEM_BLOCK 256

// log2(e) / 4 : exp(x/4) == exp2(x * S)
#define CEM_S 0.360673760222240781f

#if defined(__has_builtin)
#if __has_builtin(__builtin_amdgcn_exp2f)
#define CEM_EXP2(x) __builtin_amdgcn_exp2f(x)
#endif
#if __has_builtin(__builtin_amdgcn_global_load_async_to_lds_b128) && \
    __has_builtin(__builtin_amdgcn_s_wait_asynccnt)
#define CEM_USE_ASYNC_LDS 1
#endif
#endif
#ifndef CEM_EXP2
#define CEM_EXP2(x) exp2f(x)
#endif

// Vector types: ext_vector_type works with __builtin_nontemporal_load
// (HIP's float4 is a class type and does not).
typedef float cem_v4f __attribute__((ext_vector_type(4)));
typedef int cem_v4i __attribute__((vector_size(4 * sizeof(int))));
typedef __attribute__((address_space(1))) cem_v4i* cem_gptr;  // global
typedef __attribute__((address_space(3))) cem_v4i* cem_lptr;  // LDS

// ---------------- Kernel 1: one block per row ----------------
__global__ __launch_bounds__(CEM_BLOCK) void cem_row_kernel(
    const float* __restrict__ logits, const int* __restrict__ label,
    float* __restrict__ partial) {
  const int row = blockIdx.x;
  const int tid = threadIdx.x;

  const cem_v4f* __restrict__ rowv =
      reinterpret_cast<const cem_v4f*>(logits + (size_t)row * CEM_CLASSES);

  // 4 independent accumulators hide v_exp_f32 latency; non-temporal loads:
  // every logit is touched exactly once, so don't churn L2 allocation.
  float s0 = 0.f, s1 = 0.f, s2 = 0.f, s3 = 0.f;
#pragma unroll 2
  for (int i = tid; i < CEM_CLASSES4; i += CEM_BLOCK) {
    cem_v4f v = __builtin_nontemporal_load(rowv + i);
    s0 += CEM_EXP2(v.x * CEM_S);
    s1 += CEM_EXP2(v.y * CEM_S);
    s2 += CEM_EXP2(v.z * CEM_S);
    s3 += CEM_EXP2(v.w * CEM_S);
  }
  float s = (s0 + s1) + (s2 + s3);

  __shared__ float red[CEM_BLOCK];
  red[tid] = s;
  __syncthreads();
#pragma unroll
  for (int off = CEM_BLOCK / 2; off > 0; off >>= 1) {
    if (tid < off) red[tid] += red[tid + off];
    __syncthreads();
  }

  if (tid == 0) {
    const int lab = label[row];
    const float ly = logits[(size_t)row * CEM_CLASSES + lab];
    const float num = expf(4.0f * ly);
    const float denom = red[0] - expf(0.25f * ly) + num;
    // log(num/denom) = 4*ly - log(denom)
    partial[row] = 4.0f * ly - logf(denom);
  }
}

// ---------------- Kernel 2: final reduction over 8192 partials ----------------
__global__ __launch_bounds__(CEM_BLOCK) void cem_final_kernel(
    const float* __restrict__ partial, float* __restrict__ out) {
  __shared__ float buf[CEM_BATCH];   // 32 KB staged in LDS
  __shared__ float red[CEM_BLOCK];
  const int tid = threadIdx.x;

#if defined(CEM_USE_ASYNC_LDS)
  // gfx1250 async load engine: each lane DMAs 16B global -> LDS, tracked by
  // ASYNCcnt; wait own wave's counter, then barrier so all waves' data is
  // visible.
#pragma unroll
  for (int i = 0; i < CEM_BATCH / 4 / CEM_BLOCK; ++i) {
    const int v4 = tid + i * CEM_BLOCK;  // float4 index
    __builtin_amdgcn_global_load_async_to_lds_b128(
        (cem_gptr)(partial + (size_t)v4 * 4),
        (cem_lptr)(buf + (size_t)v4 * 4),
        0, 0);
  }
  __builtin_amdgcn_s_wait_asynccnt(0);
  __syncthreads();
#else
  for (int i = tid; i < CEM_BATCH; i += CEM_BLOCK) buf[i] = partial[i];
  __syncthreads();
#endif

  float s = 0.f;
#pragma unroll 8
  for (int i = tid; i < CEM_BATCH; i += CEM_BLOCK) s += buf[i];

  red[tid] = s;
  __syncthreads();
#pragma unroll
  for (int off = CEM_BLOCK / 2; off > 0; off >>= 1) {
    if (tid < off) red[tid] += red[tid + off];
    __syncthreads();
  }
  if (tid == 0) out[0] = -(red[0] * (1.0f / (float)CEM_BATCH));
}

extern "C" void kernel_launch(void* const* d_in, const int* in_sizes, int n_in,
                              void* d_out, int out_size, void* d_ws, size_t ws_size,
                              hipStream_t stream) {
  const float* logits = (const float*)d_in[0];
  const int* label = (const int*)d_in[1];
  float* partial = (float*)d_ws;  // 8192 floats = 32 KB scratch
  float* out = (float*)d_out;

  cem_row_kernel<<<CEM_BATCH, CEM_BLOCK, 0, stream>>>(logits, label, partial);
  cem_final_kernel<<<1, CEM_BLOCK, 0, stream>>>(partial, out);
}